// RationalKANLayer_10599979287093
// MI455X (gfx1250) — compile-verified
//
#include <hip/hip_runtime.h>

typedef __attribute__((ext_vector_type(2))) float v2f;
typedef __attribute__((ext_vector_type(8))) float v8f;

#define IN_F   256
#define OUT_F  512
#define BATCH  2048

// d_ws layout:
//   [0, 4MB)        swizzled P/Q coefs: float4 per (otile, i, lane)   32*256*32*16B
//   [4MB, 4.5MB)    swizzled W:         float2 per (otile, i4, lane)  32*64*32*8B
#define SWZ_COEF_BYTES (32ull * 256ull * 32ull * 16ull)

// ---------------------------------------------------------------------------
// Prologue 1: put P/Q coefficients into the exact 16x16x4 WMMA B-operand
// lane layout.  B(4x16) layout: lane n (0..15) holds rows K=0 (v0), K=1 (v1);
// lane n+16 holds K=2 (v0), K=3 (v1).  Q's K=0 row is the constant 1+1e-6
// (folds the "1 + ..." and the eps of the reference into the WMMA).
// ---------------------------------------------------------------------------
__global__ __launch_bounds__(256) void swz_coef_kernel(
    const float* __restrict__ P, const float* __restrict__ Q,
    float4* __restrict__ dst) {
  unsigned t    = blockIdx.x * 256u + threadIdx.x;     // 262144 threads
  unsigned lane = t & 31u;
  unsigned i    = (t >> 5) & 255u;
  unsigned ot   = t >> 13;
  unsigned o    = ot * 16u + (lane & 15u);
  const float4 pc = *(const float4*)(P + ((size_t)o * IN_F + i) * 4u);
  const float* qr = Q + ((size_t)o * IN_F + i) * 3u;
  float4 v;
  if (lane < 16u) { v.x = pc.x; v.y = pc.y; v.z = 1.0f + 1e-6f; v.w = qr[0]; }
  else            { v.x = pc.z; v.y = pc.w; v.z = qr[1];        v.w = qr[2]; }
  dst[t] = v;
}

// Prologue 2: swizzle base_weight rows into WMMA B layout per 4-wide K chunk.
__global__ __launch_bounds__(256) void swz_w_kernel(
    const float* __restrict__ W, float2* __restrict__ dst) {
  unsigned t    = blockIdx.x * 256u + threadIdx.x;     // 65536 threads
  unsigned lane = t & 31u;
  unsigned i4   = (t >> 5) & 63u;
  unsigned ot   = t >> 11;
  unsigned o    = ot * 16u + (lane & 15u);
  unsigned ib   = i4 * 4u + ((lane & 16u) ? 2u : 0u);
  dst[t] = *(const float2*)(W + (size_t)o * IN_F + ib);  // 8B aligned (ib even)
}

// ---------------------------------------------------------------------------
// Main kernel. Block = 256 threads = 8 waves. Block covers 32 batches x 128
// outs; each wave owns one 16-wide o-tile and two 16-batch C tiles (sharing
// B operands).  Per i: 1 coalesced b128 coef load + 2 ds loads + 4 WMMAs +
// 16x(rcp + NR + fma).  Base GEMM folded in as 2 WMMAs per 4 i-steps.
// ---------------------------------------------------------------------------
__global__ __launch_bounds__(256) void rkan_main(
    const float* __restrict__ x, const float* __restrict__ bias,
    const float4* __restrict__ swzC, const float2* __restrict__ swzW,
    float* __restrict__ out) {
  // stride 260: row base 1040B (16B aligned), banks (4m + i) % 64 conflict-free
  __shared__ float xs[32][260];

  const unsigned tid  = threadIdx.x;
  const unsigned lane = tid & 31u;
  const unsigned wave = tid >> 5;
  const unsigned ml   = lane & 15u;
  const bool     hiv  = (lane >= 16u);
  const unsigned b0   = blockIdx.y * 32u;
  const unsigned ot   = blockIdx.x * 8u + wave;
  const unsigned o0   = ot * 16u;

  // Cooperative, coalesced load of 32x256 x-tile into padded LDS.
  const float4* xsrc = (const float4*)(x + (size_t)b0 * IN_F);
#pragma unroll
  for (int j = 0; j < 8; ++j) {
    unsigned f   = tid + 256u * (unsigned)j;     // float4 index, 0..2047
    float4   v   = xsrc[f];
    unsigned row = f >> 6;
    unsigned col = (f & 63u) * 4u;
    *(float4*)&xs[row][col] = v;
  }
  __syncthreads();

  v8f acc0 = {};
  v8f acc1 = {};
  const v8f zero = {};
  const float4* cp = swzC + ((size_t)ot * 256u * 32u + lane);
  const float2* wp = swzW + ((size_t)ot * 64u * 32u + lane);

  for (int i4 = 0; i4 < 64; ++i4) {
    // ---- base GEMM: acc += x[:, i4*4 .. i4*4+3] * W-chunk ----
    {
      unsigned ib = (unsigned)i4 * 4u + (hiv ? 2u : 0u);
      float2 t0 = *(const float2*)&xs[ml][ib];        // 8B aligned
      float2 t1 = *(const float2*)&xs[ml + 16u][ib];
      float2 wv = wp[(size_t)i4 * 32u];
      v2f xa0; xa0.x = t0.x; xa0.y = t0.y;
      v2f xa1; xa1.x = t1.x; xa1.y = t1.y;
      v2f bw;  bw.x  = wv.x; bw.y  = wv.y;
      acc0 = __builtin_amdgcn_wmma_f32_16x16x4_f32(false, xa0, false, bw,
                                                   (short)0, acc0, false, false);
      acc1 = __builtin_amdgcn_wmma_f32_16x16x4_f32(false, xa1, false, bw,
                                                   (short)0, acc1, false, false);
    }

    // ---- rational part, 4 i-values ----
#pragma unroll
    for (int j2 = 0; j2 < 4; ++j2) {
      int   i  = i4 * 4 + j2;
      float x0 = xs[ml][i];
      float x1 = xs[ml + 16u][i];
      float4 c = cp[(size_t)i * 32u];                 // coalesced b128 (L2-hot)
      v2f bp; bp.x = c.x; bp.y = c.y;                 // P coef rows (K pair)
      v2f bq; bq.x = c.z; bq.y = c.w;                 // Q coef rows (K pair)

      // A = [1, x, x^2, x^3] in 16x4 WMMA layout
      float s0 = x0 * x0, u0 = s0 * x0;
      float s1 = x1 * x1, u1 = s1 * x1;
      v2f a0; a0.x = hiv ? s0 : 1.0f; a0.y = hiv ? u0 : x0;
      v2f a1; a1.x = hiv ? s1 : 1.0f; a1.y = hiv ? u1 : x1;

      v8f p0 = __builtin_amdgcn_wmma_f32_16x16x4_f32(false, a0, false, bp,
                                                     (short)0, zero, false, false);
      v8f q0 = __builtin_amdgcn_wmma_f32_16x16x4_f32(false, a0, false, bq,
                                                     (short)0, zero, false, false);
      v8f p1 = __builtin_amdgcn_wmma_f32_16x16x4_f32(false, a1, false, bp,
                                                     (short)0, zero, false, false);
      v8f q1 = __builtin_amdgcn_wmma_f32_16x16x4_f32(false, a1, false, bq,
                                                     (short)0, zero, false, false);

      // divide + accumulate: rcp + one Newton step + fma, per cell
#pragma unroll
      for (int j = 0; j < 8; ++j) {
        float qa = q0[j];
        float ra = __builtin_amdgcn_rcpf(qa);
        ra = fmaf(fmaf(-qa, ra, 1.0f), ra, ra);
        acc0[j] = fmaf(p0[j], ra, acc0[j]);
        float qb = q1[j];
        float rb = __builtin_amdgcn_rcpf(qb);
        rb = fmaf(fmaf(-qb, rb, 1.0f), rb, rb);
        acc1[j] = fmaf(p1[j], rb, acc1[j]);
      }
    }
  }

  // Epilogue: C layout (vgpr j -> M = j or j+8 per lane half, N = lane%16)
  float bb  = bias[o0 + ml];
  unsigned col = o0 + ml;
#pragma unroll
  for (int j = 0; j < 8; ++j) {
    unsigned r0 = b0 + (unsigned)j + (hiv ? 8u : 0u);
    out[(size_t)r0 * OUT_F + col]         = acc0[j] + bb;
    out[(size_t)(r0 + 16u) * OUT_F + col] = acc1[j] + bb;
  }
}

// ---------------------------------------------------------------------------
extern "C" void kernel_launch(void* const* d_in, const int* in_sizes, int n_in,
                              void* d_out, int out_size, void* d_ws, size_t ws_size,
                              hipStream_t stream) {
  const float* x = (const float*)d_in[0];
  const float* P = (const float*)d_in[1];
  const float* Q = (const float*)d_in[2];
  const float* W = (const float*)d_in[3];
  const float* b = (const float*)d_in[4];

  float4* swzC = (float4*)d_ws;
  float2* swzW = (float2*)((char*)d_ws + SWZ_COEF_BYTES);

  swz_coef_kernel<<<262144 / 256, 256, 0, stream>>>(P, Q, swzC);
  swz_w_kernel<<<65536 / 256, 256, 0, stream>>>(W, swzW);

  dim3 grid(OUT_F / 128, BATCH / 32);   // (4, 64) blocks, 256 threads each
  rkan_main<<<grid, 256, 0, stream>>>(x, b, swzC, swzW, (float*)d_out);
}